// TopicGraphEncoder_77077483094303
// MI455X (gfx1250) — compile-verified
//
#include <hip/hip_runtime.h>
#include <math.h>

// ---------------- problem constants (match reference) ----------------
#define N_SENT     100000
#define N_TOPIC    10000
#define N_USER     50000
#define N_ITEM     50000
#define F_DIM      128
#define F4_DIM     32          // F/4
#define E_EDGES    500000

typedef __attribute__((ext_vector_type(2))) float v2f;
typedef __attribute__((ext_vector_type(8))) float v8f;

// ---- ordered-uint encoding so atomicMax(uint) == float max ----
__device__ __forceinline__ unsigned f2key(float f) {
    unsigned u = __float_as_uint(f);
    return (u & 0x80000000u) ? ~u : (u | 0x80000000u);
}
__device__ __forceinline__ float key2f(unsigned k) {
    return (k & 0x80000000u) ? __uint_as_float(k & 0x7fffffffu)
                             : __uint_as_float(~k);
}

__device__ __forceinline__ float gelu_exact(float x) {
    return 0.5f * x * (1.0f + erff(x * 0.70710678118654752f));
}

// ---------------- node feature kernels ----------------
// h[i] = emb[sid[i]] * w1[tid[i]]   (per (node, float4-chunk))
__global__ void k_sent_h(const float4* __restrict__ emb,
                         const float4* __restrict__ w1,
                         const int* __restrict__ sid,
                         const int* __restrict__ tid,
                         float4* __restrict__ h, int n) {
    int idx = blockIdx.x * blockDim.x + threadIdx.x;
    if (idx >= n * F4_DIM) return;
    int i = idx >> 5, f = idx & 31;
    float4 a = emb[(size_t)sid[i] * F4_DIM + f];
    float4 b = w1[(size_t)tid[i] * F4_DIM + f];
    float4 r; r.x = a.x * b.x; r.y = a.y * b.y; r.z = a.z * b.z; r.w = a.w * b.w;
    h[(size_t)i * F4_DIM + f] = r;
}

// out[i] = gelu(feat[i] * w[tid[i]])
__global__ void k_gate(const float4* __restrict__ feat,
                       const float4* __restrict__ w,
                       const int* __restrict__ tid,
                       float4* __restrict__ out, int n) {
    int idx = blockIdx.x * blockDim.x + threadIdx.x;
    if (idx >= n * F4_DIM) return;
    int i = idx >> 5, f = idx & 31;
    float4 a = feat[(size_t)i * F4_DIM + f];
    float4 b = w[(size_t)tid[i] * F4_DIM + f];
    float4 r;
    r.x = gelu_exact(a.x * b.x); r.y = gelu_exact(a.y * b.y);
    r.z = gelu_exact(a.z * b.z); r.w = gelu_exact(a.w * b.w);
    out[(size_t)i * F4_DIM + f] = r;
}

// ---------------- attention aggregation kernels ----------------
// sum[dst[e]] += h[src[e]]
__global__ void k_scatter_sum(const float4* __restrict__ h,
                              const int* __restrict__ src,
                              const int* __restrict__ dst,
                              float* __restrict__ sum, int E) {
    int idx = blockIdx.x * blockDim.x + threadIdx.x;
    if (idx >= E * F4_DIM) return;
    int e = idx >> 5, f = idx & 31;
    float4 v = h[(size_t)src[e] * F4_DIM + f];
    float* p = sum + (size_t)dst[e] * F_DIM + f * 4;
    atomicAdd(p + 0, v.x); atomicAdd(p + 1, v.y);
    atomicAdd(p + 2, v.z); atomicAdd(p + 3, v.w);
}

// score[e] = dot(h[src[e]], sum[dst[e]]);  mkey[dst] = max(score)   (1 wave/edge)
__global__ void k_edge_score(const float4* __restrict__ h,
                             const float4* __restrict__ sum,
                             const int* __restrict__ src,
                             const int* __restrict__ dst,
                             float* __restrict__ score,
                             unsigned* __restrict__ mkey, int E) {
    int e = blockIdx.x * (blockDim.x >> 5) + (threadIdx.x >> 5);
    int lane = threadIdx.x & 31;
    if (e >= E) return;
    float4 a = h[(size_t)src[e] * F4_DIM + lane];
    float4 b = sum[(size_t)dst[e] * F4_DIM + lane];
    float s = a.x * b.x + a.y * b.y + a.z * b.z + a.w * b.w;
    #pragma unroll
    for (int off = 16; off > 0; off >>= 1) s += __shfl_down(s, off, 32);
    if (lane == 0) {
        score[e] = s;
        atomicMax(&mkey[dst[e]], f2key(s));
    }
}

// ex[e] = exp(score[e] - m[dst]);  z[dst] += ex[e]
__global__ void k_edge_expz(const float* __restrict__ score,
                            const int* __restrict__ dst,
                            const unsigned* __restrict__ mkey,
                            float* __restrict__ ex,
                            float* __restrict__ z, int E) {
    int e = blockIdx.x * blockDim.x + threadIdx.x;
    if (e >= E) return;
    float m = key2f(mkey[dst[e]]);
    float v = expf(score[e] - m);
    ex[e] = v;
    atomicAdd(&z[dst[e]], v);
}

// out[dst[e]] += h[src[e]] * (ex[e]/z[dst[e]])
__global__ void k_scatter_wsum(const float4* __restrict__ h,
                               const int* __restrict__ src,
                               const int* __restrict__ dst,
                               const float* __restrict__ ex,
                               const float* __restrict__ z,
                               float* __restrict__ out, int E) {
    int idx = blockIdx.x * blockDim.x + threadIdx.x;
    if (idx >= E * F4_DIM) return;
    int e = idx >> 5, f = idx & 31;
    int d = dst[e];
    float a = ex[e] / z[d];
    float4 v = h[(size_t)src[e] * F4_DIM + f];
    float* p = out + (size_t)d * F_DIM + f * 4;
    atomicAdd(p + 0, v.x * a); atomicAdd(p + 1, v.y * a);
    atomicAdd(p + 2, v.z * a); atomicAdd(p + 3, v.w * a);
}

// ---------------- WMMA f32 GEMM: out = A @ W^T + bias ----------------
// A:[M,128] row-major, W:[128,128] row-major, out:[M,128].
// One wave computes one 16x16 output tile; K loop of 32 x V_WMMA_F32_16X16X4_F32.
// A layout (ISA 7.12.2, 32-bit A 16x4): lanes 0-15 -> {K, K+1}; lanes 16-31 -> {K+2, K+3}.
// B (4x16 = W^T tile): B[k][n] = W[n][k]; mirrored lane layout.
// C/D layout: VGPR r, lanes 0-15 -> row r; lanes 16-31 -> row r+8.
__global__ void k_gemm_bias_wmma(const float* __restrict__ A,
                                 const float* __restrict__ W,
                                 const float* __restrict__ bias,
                                 float* __restrict__ out, int M) {
    const int lane = threadIdx.x;            // 0..31
    const int half = lane >> 4;              // 0 | 1
    const int l    = lane & 15;
    const int n0   = threadIdx.y * 16;       // 8 waves cover N = 0..127
    const int m0   = blockIdx.x * 16;

    const float* __restrict__ arow = A + (size_t)(m0 + l) * F_DIM;
    const float* __restrict__ wrow = W + (size_t)(n0 + l) * F_DIM;

    v8f c = {};
    #pragma unroll
    for (int k = 0; k < F_DIM; k += 4) {
        const int ka = k + half * 2;
        v2f a, b;
        a[0] = arow[ka]; a[1] = arow[ka + 1];
        b[0] = wrow[ka]; b[1] = wrow[ka + 1];
        c = __builtin_amdgcn_wmma_f32_16x16x4_f32(
                /*neg_a=*/false, a, /*neg_b=*/false, b,
                /*c_mod=*/(short)0, c, /*reuse_a=*/false, /*reuse_b=*/false);
    }

    const int col = n0 + l;
    const float bv = bias[col];
    #pragma unroll
    for (int r = 0; r < 8; ++r) {
        const int row = m0 + r + half * 8;
        out[(size_t)row * F_DIM + col] = c[r] + bv;
    }
}

// ---------------- host-side orchestration ----------------
static inline int cdiv(int a, int b) { return (a + b - 1) / b; }

static void run_attn(const float* h, const int* src, const int* dst, int E, int n_dst,
                     float* sum, float* score, float* ex, unsigned* mkey, float* z,
                     float* agg, hipStream_t s) {
    hipMemsetAsync(sum,  0, (size_t)n_dst * F_DIM * sizeof(float), s);
    hipMemsetAsync(agg,  0, (size_t)n_dst * F_DIM * sizeof(float), s);
    hipMemsetAsync(mkey, 0, (size_t)n_dst * sizeof(unsigned), s);
    hipMemsetAsync(z,    0, (size_t)n_dst * sizeof(float), s);

    int t = 256;
    k_scatter_sum<<<cdiv(E * F4_DIM, t), t, 0, s>>>((const float4*)h, src, dst, sum, E);
    k_edge_score<<<cdiv(E, t / 32), t, 0, s>>>((const float4*)h, (const float4*)sum,
                                               src, dst, score, mkey, E);
    k_edge_expz<<<cdiv(E, t), t, 0, s>>>(score, dst, mkey, ex, z, E);
    k_scatter_wsum<<<cdiv(E * F4_DIM, t), t, 0, s>>>((const float4*)h, src, dst, ex, z, agg, E);
}

extern "C" void kernel_launch(void* const* d_in, const int* in_sizes, int n_in,
                              void* d_out, int out_size, void* d_ws, size_t ws_size,
                              hipStream_t stream) {
    // inputs (setup_inputs order)
    const int* sentence_id  = (const int*)d_in[0];
    const int* st_topic_id  = (const int*)d_in[1];
    const int* st_src       = (const int*)d_in[2];
    const int* st_dst       = (const int*)d_in[3];
    const int* tu_topic_id  = (const int*)d_in[4];
    const int* tu_src       = (const int*)d_in[5];
    const int* tu_dst       = (const int*)d_in[6];
    const int* ti_topic_id  = (const int*)d_in[7];
    const int* ti_src       = (const int*)d_in[8];
    const int* ti_dst       = (const int*)d_in[9];
    const float* emb          = (const float*)d_in[10];
    const float* sentence_w1  = (const float*)d_in[11];
    const float* w_sl         = (const float*)d_in[12];
    const float* b_sl         = (const float*)d_in[13];
    const float* topic_user_w = (const float*)d_in[14];
    const float* w_ul         = (const float*)d_in[15];
    const float* b_ul         = (const float*)d_in[16];
    const float* topic_item_w = (const float*)d_in[17];
    const float* w_il         = (const float*)d_in[18];
    const float* b_il         = (const float*)d_in[19];

    float* out = (float*)d_out;
    float* uo = out;                                 // [N_USER, F]
    float* io = out + (size_t)N_USER * F_DIM;        // [N_ITEM, F]

    // workspace carve (floats)
    float* ws = (float*)d_ws;
    float*    buf_h      = ws;                                   size_t o = (size_t)N_SENT * F_DIM;
    float*    buf_sum    = ws + o;                               o += (size_t)N_USER * F_DIM;
    float*    buf_agg    = ws + o;                               o += (size_t)N_USER * F_DIM;
    float*    topic_feat = ws + o;                               o += (size_t)N_TOPIC * F_DIM;
    float*    score      = ws + o;                               o += E_EDGES;
    float*    ex         = ws + o;                               o += E_EDGES;
    float*    z          = ws + o;                               o += N_USER;
    unsigned* mkey       = (unsigned*)(ws + o);                  o += N_USER;
    (void)ws_size; (void)n_in; (void)in_sizes; (void)out_size;

    const int t = 256;

    // ---- stage 1: sentence -> topic ----
    k_sent_h<<<cdiv(N_SENT * F4_DIM, t), t, 0, stream>>>(
        (const float4*)emb, (const float4*)sentence_w1, sentence_id, st_topic_id,
        (float4*)buf_h, N_SENT);
    run_attn(buf_h, st_src, st_dst, E_EDGES, N_TOPIC,
             buf_sum, score, ex, mkey, z, buf_agg, stream);
    // topic_feat = topic_h @ w_sl^T + b_sl
    k_gemm_bias_wmma<<<dim3(N_TOPIC / 16), dim3(32, 8), 0, stream>>>(
        buf_agg, w_sl, b_sl, topic_feat, N_TOPIC);

    // ---- stage 2: topic -> user ----
    k_gate<<<cdiv(N_TOPIC * F4_DIM, t), t, 0, stream>>>(
        (const float4*)topic_feat, (const float4*)topic_user_w, tu_topic_id,
        (float4*)buf_h, N_TOPIC);
    run_attn(buf_h, tu_src, tu_dst, E_EDGES, N_USER,
             buf_sum, score, ex, mkey, z, buf_agg, stream);
    k_gemm_bias_wmma<<<dim3(N_USER / 16), dim3(32, 8), 0, stream>>>(
        buf_agg, w_ul, b_ul, uo, N_USER);

    // ---- stage 3: topic -> item ----
    k_gate<<<cdiv(N_TOPIC * F4_DIM, t), t, 0, stream>>>(
        (const float4*)topic_feat, (const float4*)topic_item_w, ti_topic_id,
        (float4*)buf_h, N_TOPIC);
    run_attn(buf_h, ti_src, ti_dst, E_EDGES, N_ITEM,
             buf_sum, score, ex, mkey, z, buf_agg, stream);
    k_gemm_bias_wmma<<<dim3(N_ITEM / 16), dim3(32, 8), 0, stream>>>(
        buf_agg, w_il, b_il, io, N_ITEM);
}